// NeuralCube_46084999086188
// MI455X (gfx1250) — compile-verified
//
#include <hip/hip_runtime.h>
#include <hip/hip_bf16.h>

typedef float v2f __attribute__((ext_vector_type(2)));
typedef float v8f __attribute__((ext_vector_type(8)));

#define CUBE   32
#define PITCH  34            // even pitch: 8B-aligned rows, conflict-free b64 LDS access
#define DIN    784
#define NSURF  1024          // CUBE*CUBE
#define NCELL  32768         // CUBE^3
#define DOUT   10
#define ALPHA  0.5f

#if __has_builtin(__builtin_amdgcn_tanhf)
__device__ __forceinline__ float fast_tanh(float x) { return __builtin_amdgcn_tanhf(x); }
#else
__device__ __forceinline__ float fast_tanh(float x) { return tanhf(x); }
#endif

// ---------------------------------------------------------------------------
// Kernel 1: surface = x @ W_in^T + b_in  (1024 x 1024, K = 784), fp32 WMMA.
// One wave per 16x16 C-tile, K-loop of V_WMMA_F32_16X16X4_F32.
//   A fragment (16x4 f32): lane<16 -> M=lane, K=k,k+1 ; lane>=16 -> M=lane-16, K=k+2,k+3
//   B fragment (4x16 f32): same pattern on W_in rows (B = W_in^T, so B[k][n] = W_in[n][k])
//   D (16x16 f32, 8 VGPRs): VGPR j -> M = j + 8*(lane>=16), N = lane&15
// ---------------------------------------------------------------------------
__global__ __launch_bounds__(256)
void surface_gemm_wmma(const float* __restrict__ x,
                       const float* __restrict__ W_in,
                       const float* __restrict__ b_in,
                       float* __restrict__ surface)
{
    const int wave  = blockIdx.x * (blockDim.x >> 5) + (threadIdx.x >> 5);
    const int lane  = threadIdx.x & 31;
    const int tileM = wave >> 6;          // 64 tiles along N (NSURF/16)
    const int tileN = wave & 63;
    const int m0    = lane & 15;
    const int koff  = (lane >> 4) << 1;   // 0 for lanes 0-15, 2 for lanes 16-31

    const float* arow = x    + (size_t)(tileM * 16 + m0) * DIN;
    const float* brow = W_in + (size_t)(tileN * 16 + m0) * DIN;

    v8f c = {};
    #pragma unroll 4
    for (int k = 0; k < DIN; k += 4) {
        v2f a = *(const v2f*)(arow + k + koff);
        v2f b = *(const v2f*)(brow + k + koff);
        c = __builtin_amdgcn_wmma_f32_16x16x4_f32(
                /*neg_a=*/false, a, /*neg_b=*/false, b,
                /*c_mod=*/(short)0, c, /*reuse_a=*/false, /*reuse_b=*/false);
    }

    const int col     = tileN * 16 + m0;
    const int rowBase = tileM * 16 + ((lane >> 4) << 3);
    const float bias  = b_in[col];
    #pragma unroll
    for (int j = 0; j < 8; ++j)
        surface[(size_t)(rowBase + j) * NSURF + col] = c[j] + bias;
}

// ---------------------------------------------------------------------------
// Kernel 2: per-batch-item cube evolution, state resident in LDS (136 KB << 320 KB/WGP).
// Block = 1024 threads (32 waves); thread t owns row (z = t>>5, y = t&31, x=0..31).
// 30x { conv3x3x3: 8 neighbor rows from LDS (b64, conflict-free) + own row from
// registers + native tanh + EMA }, then the 10-way output projection reduced
// in-wave (shfl) + ds_add_f32.
// ---------------------------------------------------------------------------
__global__ __launch_bounds__(1024)
void cube_evolve(const float* __restrict__ surface,
                 const float* __restrict__ conv_w,
                 const float* __restrict__ neuron_bias,
                 const float* __restrict__ W_out,
                 const float* __restrict__ b_out,
                 const int*   __restrict__ steps_ptr,
                 float* __restrict__ out)
{
    extern __shared__ float smem[];
    float* S   = smem;                       // [CUBE][CUBE][PITCH]
    float* red = smem + CUBE * CUBE * PITCH; // [DOUT]

    const int b = blockIdx.x;
    const int t = threadIdx.x;               // 0..1023
    const int z = t >> 5;
    const int y = t & 31;
    const int steps = *steps_ptr;

    // conv weights: uniform across the grid -> scalar loads
    float w[27];
    #pragma unroll
    for (int i = 0; i < 27; ++i) w[i] = conv_w[i];

    // per-cell constant drive: neuron_bias (+ injected input face at z==0)
    float base[CUBE];
    {
        const float* nb = neuron_bias + (size_t)t * CUBE;
        const float* sf = surface + (size_t)b * NSURF + (size_t)y * CUBE;
        #pragma unroll
        for (int xx = 0; xx < CUBE; ++xx) {
            float v = nb[xx];
            if (z == 0) v += sf[xx];
            base[xx] = v;
        }
    }

    float* myrow = S + (z * CUBE + y) * PITCH;   // 8B-aligned (PITCH even)
    {
        float2* m2 = (float2*)myrow;
        #pragma unroll
        for (int i = 0; i < CUBE / 2; ++i) m2[i] = make_float2(0.0f, 0.0f);
    }

    float cur[CUBE];                          // this thread's row of the state
    #pragma unroll
    for (int xx = 0; xx < CUBE; ++xx) cur[xx] = 0.0f;

    __syncthreads();

    for (int s = 0; s < steps; ++s) {
        float acc[CUBE];
        #pragma unroll
        for (int xx = 0; xx < CUBE; ++xx) acc[xx] = base[xx];

        #pragma unroll
        for (int dz = -1; dz <= 1; ++dz) {
            const int zz = z + dz;
            if (zz < 0 || zz >= CUBE) continue;
            #pragma unroll
            for (int dy = -1; dy <= 1; ++dy) {
                const int yy = y + dy;
                if (yy < 0 || yy >= CUBE) continue;
                const float w0 = w[(dz + 1) * 9 + (dy + 1) * 3 + 0];
                const float w1 = w[(dz + 1) * 9 + (dy + 1) * 3 + 1];
                const float w2 = w[(dz + 1) * 9 + (dy + 1) * 3 + 2];

                float rr[CUBE];
                if (dz == 0 && dy == 0) {
                    // center row: already live in registers, skip the LDS read
                    #pragma unroll
                    for (int xx = 0; xx < CUBE; ++xx) rr[xx] = cur[xx];
                } else {
                    const float2* r2 = (const float2*)(S + (zz * CUBE + yy) * PITCH);
                    #pragma unroll
                    for (int i = 0; i < CUBE / 2; ++i) {     // ds_load_b64, conflict-free
                        float2 v = r2[i];
                        rr[2 * i]     = v.x;
                        rr[2 * i + 1] = v.y;
                    }
                }
                #pragma unroll
                for (int xx = 0; xx < CUBE; ++xx) {
                    float a = acc[xx] + w1 * rr[xx];
                    if (xx > 0)        a += w0 * rr[xx - 1];   // zero pad at x=0
                    if (xx < CUBE - 1) a += w2 * rr[xx + 1];   // zero pad at x=31
                    acc[xx] = a;
                }
            }
        }

        // state' = (1-ALPHA)*state + ALPHA*tanh(conv + x_cube + neuron_bias)
        #pragma unroll
        for (int xx = 0; xx < CUBE; ++xx)
            cur[xx] = (1.0f - ALPHA) * cur[xx] + ALPHA * fast_tanh(acc[xx]);

        __syncthreads();                       // everyone done reading old S
        {
            float2* m2 = (float2*)myrow;
            #pragma unroll
            for (int i = 0; i < CUBE / 2; ++i)  // ds_store_b64
                m2[i] = make_float2(cur[2 * i], cur[2 * i + 1]);
        }
        __syncthreads();                       // new S visible
    }

    // out[b] = state_flat @ W_out^T + b_out ; cur[] already holds final state.
    if (t < DOUT) red[t] = 0.0f;
    __syncthreads();

    #pragma unroll
    for (int o = 0; o < DOUT; ++o) {
        const float4* wr = (const float4*)(W_out + (size_t)o * NCELL + (size_t)t * CUBE);
        float p = 0.0f;
        #pragma unroll
        for (int i = 0; i < CUBE / 4; ++i) {    // global_load_b128
            float4 v = wr[i];
            p += cur[4 * i]     * v.x;
            p += cur[4 * i + 1] * v.y;
            p += cur[4 * i + 2] * v.z;
            p += cur[4 * i + 3] * v.w;
        }
        #pragma unroll
        for (int off = 16; off > 0; off >>= 1)
            p += __shfl_down(p, off, 32);
        if ((t & 31) == 0) atomicAdd(&red[o], p);   // ds_add_f32
    }
    __syncthreads();
    if (t < DOUT) out[(size_t)b * DOUT + t] = red[t] + b_out[t];
}

// ---------------------------------------------------------------------------
extern "C" void kernel_launch(void* const* d_in, const int* in_sizes, int n_in,
                              void* d_out, int out_size, void* d_ws, size_t ws_size,
                              hipStream_t stream)
{
    const float* x           = (const float*)d_in[0];
    const float* W_in        = (const float*)d_in[1];
    const float* b_in        = (const float*)d_in[2];
    const float* conv_w      = (const float*)d_in[3];
    const float* neuron_bias = (const float*)d_in[4];
    const float* W_out       = (const float*)d_in[5];
    const float* b_out       = (const float*)d_in[6];
    const int*   steps       = (const int*)d_in[7];
    float*       outp        = (float*)d_out;

    const int B = in_sizes[0] / DIN;          // 1024
    float* surface = (float*)d_ws;            // B * NSURF floats (4 MB)

    // GEMM 1: one wave per 16x16 tile -> (B/16)*(NSURF/16) waves, 8 waves/block
    const int tiles  = (B / 16) * (NSURF / 16);
    const int blocks = tiles / 8;
    surface_gemm_wmma<<<blocks, 256, 0, stream>>>(x, W_in, b_in, surface);

    // Evolution + output projection: one block per batch item.
    const size_t lds_bytes = (size_t)(CUBE * CUBE * PITCH + 16) * sizeof(float);
    cube_evolve<<<B, 1024, lds_bytes, stream>>>(surface, conv_w, neuron_bias,
                                                W_out, b_out, steps, outp);
}